// AdvectionLayer_7584912245413
// MI455X (gfx1250) — compile-verified
//
#include <hip/hip_runtime.h>
#include <cstdint>

#define BDIM  256
#define LOGW  10
#define IW    1024
#define IH    1024
#define LOGHW 20
#define HWPX  (1u << 20)

typedef __attribute__((address_space(1))) int g_int_t;   // global ("__device__ int*")
typedef __attribute__((address_space(3))) int l_int_t;   // LDS

__global__ __launch_bounds__(BDIM) void advect_warp_kernel(
    const float* __restrict__ image,   // [B,4,H,W]
    const float* __restrict__ flow,    // [B,2,H,W]
    float* __restrict__ out,           // [B,4,H,W]
    int npix)                          // B*H*W
{
    __shared__ float s_flow[2 * BDIM];

    const int tid = threadIdx.x;
    const int idx = blockIdx.x * BDIM + tid;
    if (idx >= npix) return;

    const int b   = idx >> LOGHW;
    const int rem = idx & (HWPX - 1);

    // ---- async-stage streaming flow reads into LDS (CDNA5 async path) ----
    const size_t fbase = ((size_t)b << (LOGHW + 1)) + (unsigned)rem;
#if __has_builtin(__builtin_amdgcn_global_load_async_to_lds_b32)
    __builtin_amdgcn_global_load_async_to_lds_b32(
        (g_int_t*)(flow + fbase),
        (l_int_t*)&s_flow[tid], 0, 0);
    __builtin_amdgcn_global_load_async_to_lds_b32(
        (g_int_t*)(flow + fbase + HWPX),
        (l_int_t*)&s_flow[BDIM + tid], 0, 0);
#else
    {
        unsigned l0 = (unsigned)(uintptr_t)&s_flow[tid];
        unsigned l1 = (unsigned)(uintptr_t)&s_flow[BDIM + tid];
        unsigned long long g0 = (unsigned long long)(uintptr_t)(flow + fbase);
        unsigned long long g1 = (unsigned long long)(uintptr_t)(flow + fbase + HWPX);
        asm volatile("global_load_async_to_lds_b32 %0, %1, off"
                     :: "v"(l0), "v"(g0) : "memory");
        asm volatile("global_load_async_to_lds_b32 %0, %1, off"
                     :: "v"(l1), "v"(g1) : "memory");
    }
#endif
#if __has_builtin(__builtin_amdgcn_s_wait_asynccnt)
    __builtin_amdgcn_s_wait_asynccnt(0);
#else
    asm volatile("s_wait_asynccnt 0" ::: "memory");
#endif

    const float fx = s_flow[tid];
    const float fy = s_flow[BDIM + tid];

    const int y = rem >> LOGW;
    const int x = rem & (IW - 1);

    // ---- coordinate math, mirroring the reference op-for-op (f32) ----
    const float vx = (float)x + fx;
    const float vy = (float)y + fy;
    const float gx = 2.0f * vx / 1023.0f - 1.0f;      // align_corners=True normalize
    const float gy = 2.0f * vy / 1023.0f - 1.0f;
    float ixf = ((gx + 1.0f) * 1024.0f - 1.0f) * 0.5f; // align_corners=False unnormalize
    float iyf = ((gy + 1.0f) * 1024.0f - 1.0f) * 0.5f;
    ixf = fminf(fmaxf(ixf, 0.0f), 1023.0f);            // border clamp
    iyf = fminf(fmaxf(iyf, 0.0f), 1023.0f);

    const float x0f = floorf(ixf);
    const float y0f = floorf(iyf);
    const int x0 = (int)x0f;
    const int y0 = (int)y0f;
    const float wx = ixf - x0f;
    const float wy = iyf - y0f;
    const int x1 = min(x0 + 1, IW - 1);               // mode='nearest' index clamp
    const int y1 = min(y0 + 1, IH - 1);

    const float w00 = (1.0f - wy) * (1.0f - wx);
    const float w01 = (1.0f - wy) * wx;
    const float w10 = wy * (1.0f - wx);
    const float w11 = wy * wx;

    const int r0 = y0 << LOGW;
    const int r1 = y1 << LOGW;
    const size_t bbase = ((size_t)(b << 2)) << LOGHW;

#pragma unroll
    for (int c = 0; c < 4; ++c) {
        const float* plane = image + bbase + ((size_t)c << LOGHW);
        // image gathers stay regular-temporal: 128 MB of planes fit in 192 MB L2
        const float v00 = plane[r0 + x0];
        const float v01 = plane[r0 + x1];
        const float v10 = plane[r1 + x0];
        const float v11 = plane[r1 + x1];
        const float val = v00 * w00 + v01 * w01 + v10 * w10 + v11 * w11;
        // write-once output: non-temporal store, don't evict the image from L2
        __builtin_nontemporal_store(val, out + bbase + ((size_t)c << LOGHW) + rem);
    }
}

extern "C" void kernel_launch(void* const* d_in, const int* in_sizes, int n_in,
                              void* d_out, int out_size, void* d_ws, size_t ws_size,
                              hipStream_t stream) {
    const float* image = (const float*)d_in[0];  // [8,4,1024,1024] f32
    const float* flow  = (const float*)d_in[1];  // [8,2,1024,1024] f32
    float* out = (float*)d_out;                  // [8,4,1024,1024] f32

    const int npix = in_sizes[1] / 2;            // B*H*W = 8,388,608
    const int blocks = (npix + BDIM - 1) / BDIM; // 32768 blocks of 8 wave32s
    advect_warp_kernel<<<blocks, BDIM, 0, stream>>>(image, flow, out, npix);
}